// GNN_3676492005930
// MI455X (gfx1250) — compile-verified
//
#include <hip/hip_runtime.h>

typedef __attribute__((ext_vector_type(2))) float v2f;
typedef __attribute__((ext_vector_type(8))) float v8f;

static constexpr int HF = 32;          // hidden width H
static constexpr float SLOPE = 0.2f;   // leaky-relu slope

// ---------- ordered-uint encoding for exact float atomicMax ----------
__device__ __forceinline__ unsigned fkey(float f) {
  unsigned u = __float_as_uint(f);
  return (u & 0x80000000u) ? ~u : (u | 0x80000000u);
}
__device__ __forceinline__ float funkey(unsigned u) {
  return __uint_as_float((u & 0x80000000u) ? (u ^ 0x80000000u) : ~u);
}

__global__ void fill_u32_kernel(unsigned* __restrict__ p, unsigned v, int n) {
  int i = blockIdx.x * blockDim.x + threadIdx.x;
  if (i < n) p[i] = v;
}

// ------------------------------------------------------------------
// C[M,32] += A[M,K] @ B[K,32]  (fp32, V_WMMA_F32_16X16X4_F32)
// grid.x: M-tile groups (8 waves/block, 1 tile/wave, both 16-col N-tiles
//         per wave so A is streamed from HBM exactly once)
// grid.y: K-split (partials accumulated with native fp32 atomics; C pre-zeroed)
// Fragments (ISA 7.12.2):
//   A 16x4 : lanes 0-15 -> rows, half=lane>>4 selects K pair {0,1}/{2,3}
//   B 4x16 : v0 = row k+2*half, v1 = row k+2*half+1, col = lane&15
//   C/D    : VGPR r -> row row0 + half*8 + r, col = lane&15 (+16 tile 2)
// Inner loop uses pointer increments so unrolled bodies become
// immediate-offset global loads (no per-iter 64-bit address rebuild).
// ------------------------------------------------------------------
__global__ __launch_bounds__(256) void gemm_wmma_f32(
    const float* __restrict__ A, const float* __restrict__ B,
    float* __restrict__ C, int M, int K, int kchunk) {
  const int lane = threadIdx.x & 31;
  const int wave = threadIdx.x >> 5;
  const int mtile = blockIdx.x * 8 + wave;
  const int row0 = mtile * 16;
  if (row0 >= M) return;                       // wave-uniform exit
  const int half = lane >> 4;
  const int l    = lane & 15;
  int ar = row0 + l; if (ar > M - 1) ar = M - 1;

  const int k0 = blockIdx.y * kchunk;
  int k1 = k0 + kchunk; if (k1 > K) k1 = K;
  const int steps = (k1 - k0) >> 2;            // kchunk is a multiple of 4

  // strength-reduced streaming pointers
  const float* __restrict__ ap = A + (size_t)ar * K + (k0 + 2 * half);
  const float* __restrict__ bp = B + (size_t)(k0 + 2 * half) * HF + l;

  v8f acc0 = {}; v8f acc1 = {};
  #pragma unroll 4
  for (int it = 0; it < steps; ++it) {
    v2f a = __builtin_nontemporal_load(reinterpret_cast<const v2f*>(ap)); // NT: don't pollute L2
    v2f b0, b1;
    b0.x = bp[0];       b0.y = bp[HF];
    b1.x = bp[16];      b1.y = bp[HF + 16];
    acc0 = __builtin_amdgcn_wmma_f32_16x16x4_f32(false, a, false, b0, (short)0, acc0, false, false);
    acc1 = __builtin_amdgcn_wmma_f32_16x16x4_f32(false, a, false, b1, (short)0, acc1, false, false);
    ap += 4;
    bp += 4 * HF;
  }

  const int rb = row0 + half * 8;
  #pragma unroll
  for (int r = 0; r < 8; ++r) {
    int rr = rb + r;
    if (rr < M) {
      float* crow = C + (size_t)rr * HF;
      unsafeAtomicAdd(&crow[l],      acc0[r]);
      unsafeAtomicAdd(&crow[l + 16], acc1[r]);
    }
  }
}

// s_src[i] = h[i,:]·a_src ; s_dst[i] = h[i,:]·a_dst
__global__ void node_scores_kernel(const float* __restrict__ h,
    const float* __restrict__ a_src, const float* __restrict__ a_dst,
    float* __restrict__ ss, float* __restrict__ sd, int N) {
  int i = blockIdx.x * blockDim.x + threadIdx.x;
  if (i >= N) return;
  const float* hr = h + (size_t)i * HF;
  float s0 = 0.f, s1 = 0.f;
  #pragma unroll
  for (int f = 0; f < HF; ++f) { float v = hr[f]; s0 += v * a_src[f]; s1 += v * a_dst[f]; }
  ss[i] = s0; sd[i] = s1;
}

// pass 1: e = leaky_relu(s_src[src]+s_dst[dst]); per-dst max via ordered-uint atomicMax
__global__ void edge_max_kernel(const int* __restrict__ srcI, const int* __restrict__ dstI,
    const float* __restrict__ ss, const float* __restrict__ sd,
    float* __restrict__ ev, unsigned* __restrict__ mkey, int E, int Etot) {
  int e = blockIdx.x * blockDim.x + threadIdx.x;
  if (e >= Etot) return;
  int s, d;
  if (e < E) { s = srcI[e]; d = dstI[e]; } else { s = e - E; d = s; }  // self-loops
  float v = ss[s] + sd[d];
  v = v > 0.f ? v : SLOPE * v;
  ev[e] = v;
  atomicMax(&mkey[d], fkey(v));
}

// pass 2: ex = exp(e - m[dst]); denom[dst] += ex   (ex overwrites ev in place)
__global__ void edge_expsum_kernel(const int* __restrict__ dstI,
    float* __restrict__ ev, const unsigned* __restrict__ mkey,
    float* __restrict__ denom, int E, int Etot) {
  int e = blockIdx.x * blockDim.x + threadIdx.x;
  if (e >= Etot) return;
  int d = (e < E) ? dstI[e] : (e - E);
  float x = __expf(ev[e] - funkey(mkey[d]));
  ev[e] = x;
  unsafeAtomicAdd(&denom[d], x);
}

// pass 3: agg[dst,f] += h[src,f] * ex/(denom[dst]+eps)   (lane == feature,
// so the per-edge index loads are wave-uniform and coalesce to one request)
__global__ void edge_scatter_kernel(const int* __restrict__ srcI, const int* __restrict__ dstI,
    const float* __restrict__ h, const float* __restrict__ ev, const float* __restrict__ denom,
    float* __restrict__ agg, int E, int Etot) {
  long long gid = (long long)blockIdx.x * blockDim.x + threadIdx.x;
  int e = (int)(gid >> 5);
  if (e >= Etot) return;
  int f = (int)(gid & 31);
  int s, d;
  if (e < E) { s = srcI[e]; d = dstI[e]; } else { s = e - E; d = s; }
  float alpha = ev[e] / (denom[d] + 1e-16f);
  unsafeAtomicAdd(&agg[(size_t)d * HF + f], h[(size_t)s * HF + f] * alpha);
}

// out[i,f] = relu(agg[i,f] + b[f])
__global__ void bias_relu_kernel(const float* __restrict__ agg, const float* __restrict__ b,
                                 float* __restrict__ out, int N) {
  int i = blockIdx.x * blockDim.x + threadIdx.x;
  if (i >= N * HF) return;
  float v = agg[i] + b[i & (HF - 1)];
  out[i] = v > 0.f ? v : 0.f;
}

// segment-sum over graphs
__global__ void pool_accum_kernel(const float* __restrict__ h, const int* __restrict__ batch,
    float* __restrict__ sums, float* __restrict__ cnts, int N) {
  int i = blockIdx.x * blockDim.x + threadIdx.x;
  if (i >= N * HF) return;
  int node = i >> 5, f = i & 31;
  int g = batch[node];
  unsafeAtomicAdd(&sums[(size_t)g * HF + f], h[i]);
  if (f == 0) unsafeAtomicAdd(&cnts[g], 1.0f);
}

// pooled = sums/max(cnt,1); z = relu(pooled@W1 + b1); out = z@W2 + b2
__global__ void head_kernel(const float* __restrict__ sums, const float* __restrict__ cnts,
    const float* __restrict__ w1, const float* __restrict__ b1,
    const float* __restrict__ w2, const float* __restrict__ b2,
    float* __restrict__ out, int G, int C) {
  int g = blockIdx.x * blockDim.x + threadIdx.x;
  if (g >= G) return;
  float pooled[HF], z[HF];
  float c = cnts[g]; if (c < 1.f) c = 1.f;
  #pragma unroll
  for (int f = 0; f < HF; ++f) pooled[f] = sums[(size_t)g * HF + f] / c;
  #pragma unroll
  for (int j = 0; j < HF; ++j) {
    float s = b1[j];
    #pragma unroll
    for (int k = 0; k < HF; ++k) s += pooled[k] * w1[k * HF + j];
    z[j] = s > 0.f ? s : 0.f;
  }
  for (int cc = 0; cc < C; ++cc) {
    float s = b2[cc];
    #pragma unroll
    for (int k = 0; k < HF; ++k) s += z[k] * w2[k * C + cc];
    out[(size_t)g * C + cc] = s;
  }
}

extern "C" void kernel_launch(void* const* d_in, const int* in_sizes, int n_in,
                              void* d_out, int out_size, void* d_ws, size_t ws_size,
                              hipStream_t stream) {
  const float* x     = (const float*)d_in[0];
  const int*   ei    = (const int*)d_in[1];     // [2,E] flat: src then dst
  const int*   batch = (const int*)d_in[2];
  const float* W1  = (const float*)d_in[4];
  const float* as1 = (const float*)d_in[5];
  const float* ad1 = (const float*)d_in[6];
  const float* b1  = (const float*)d_in[7];
  const float* W2  = (const float*)d_in[8];
  const float* as2 = (const float*)d_in[9];
  const float* ad2 = (const float*)d_in[10];
  const float* b2  = (const float*)d_in[11];
  const float* W3  = (const float*)d_in[12];
  const float* as3 = (const float*)d_in[13];
  const float* ad3 = (const float*)d_in[14];
  const float* b3  = (const float*)d_in[15];
  const float* l1w = (const float*)d_in[16];
  const float* l1b = (const float*)d_in[17];
  const float* l2w = (const float*)d_in[18];
  const float* l2b = (const float*)d_in[19];

  const int N    = in_sizes[2];
  const int Fin  = in_sizes[0] / N;
  const int E    = in_sizes[1] / 2;
  const int Etot = E + N;
  const int C    = in_sizes[19];
  const int G    = out_size / C;
  const int* srcI = ei;
  const int* dstI = ei + E;

  // ---- workspace carve-out (floats) ----
  float* ws = (float*)d_ws;
  size_t off = 0;
  float*    hA    = ws + off; off += (size_t)N * HF;   // GEMM output (pre-attention h)
  float*    hB    = ws + off; off += (size_t)N * HF;   // layer output / next input
  float*    agg   = ws + off; off += (size_t)N * HF;   // attention-weighted aggregation
  float*    ssrc  = ws + off; off += N;
  float*    sdst  = ws + off; off += N;
  unsigned* mkey  = (unsigned*)(ws + off); off += N;
  float*    denom = ws + off; off += N;
  float*    ev    = ws + off; off += Etot;             // edge logits, then exp values
  float*    sums  = ws + off; off += (size_t)G * HF;
  float*    cnts  = ws + off; off += G;
  (void)ws_size; (void)n_in;

  auto gat_layer = [&](const float* A, int K, const float* W,
                       const float* asv, const float* adv, const float* bv,
                       float* hout) {
    // h = A @ W  (K-split WMMA with atomic accumulation; hA pre-zeroed)
    const int mtiles = (N + 15) / 16;
    const int ksplit = (K > 256) ? 8 : 1;
    int kchunk = ((K / 4 + ksplit - 1) / ksplit) * 4;
    if (kchunk < 4) kchunk = 4;
    hipMemsetAsync(hA, 0, (size_t)N * HF * sizeof(float), stream);
    dim3 ggrid((mtiles + 7) / 8, (K + kchunk - 1) / kchunk);
    gemm_wmma_f32<<<ggrid, 256, 0, stream>>>(A, W, hA, N, K, kchunk);

    node_scores_kernel<<<(N + 255) / 256, 256, 0, stream>>>(hA, asv, adv, ssrc, sdst, N);

    hipMemsetAsync(denom, 0, (size_t)N * sizeof(float), stream);
    hipMemsetAsync(agg,   0, (size_t)N * HF * sizeof(float), stream);
    fill_u32_kernel<<<(N + 255) / 256, 256, 0, stream>>>(mkey, 0x007FFFFFu, N); // key(-inf)

    edge_max_kernel<<<(Etot + 255) / 256, 256, 0, stream>>>(srcI, dstI, ssrc, sdst, ev, mkey, E, Etot);
    edge_expsum_kernel<<<(Etot + 255) / 256, 256, 0, stream>>>(dstI, ev, mkey, denom, E, Etot);
    long long tot = (long long)Etot * HF;
    edge_scatter_kernel<<<(int)((tot + 255) / 256), 256, 0, stream>>>(srcI, dstI, hA, ev, denom, agg, E, Etot);

    bias_relu_kernel<<<(N * HF + 255) / 256, 256, 0, stream>>>(agg, bv, hout, N);
  };

  gat_layer(x,  Fin, W1, as1, ad1, b1, hB);
  gat_layer(hB, HF,  W2, as2, ad2, b2, hB);
  gat_layer(hB, HF,  W3, as3, ad3, b3, hB);

  hipMemsetAsync(sums, 0, (size_t)G * HF * sizeof(float), stream);
  hipMemsetAsync(cnts, 0, (size_t)G * sizeof(float), stream);
  pool_accum_kernel<<<(N * HF + 255) / 256, 256, 0, stream>>>(hB, batch, sums, cnts, N);
  head_kernel<<<(G + 63) / 64, 64, 0, stream>>>(sums, cnts, l1w, l1b, l2w, l2b, (float*)d_out, G, C);
}